// HighLevelFeatureExtraction_25168508354812
// MI455X (gfx1250) — compile-verified
//
#include <hip/hip_runtime.h>
#include <hip/hip_bf16.h>
#include <math.h>

// ---------------------------------------------------------------------------
// Mamba2 + MLP block stack for MI455X (gfx1250, wave32).
// All GEMMs use V_WMMA_F32_16X16X4_F32 (fp32 in/out, native for fp32 data).
// Each wave computes a 32x64 output tile (MT=2 x NT=4 WMMA tiles): 8 WMMAs
// per k-step against 6 b64 loads. __launch_bounds__(256, 1) relaxes the
// VGPR cap so all fragments of a k-step stay live in distinct registers,
// letting the scheduler stagger s_wait_loadcnt instead of serializing on a
// recycled destination pair.
// The SSM scan is parallelized over the 512 independent (b,head,p) channels:
// one wave per channel, lane n holds state h[p,n], y via __shfl_xor reduce.
// Whole working set stays L2-resident (192 MB >> ~85 MB workspace).
// ---------------------------------------------------------------------------

typedef __attribute__((ext_vector_type(2))) float v2f;
typedef __attribute__((ext_vector_type(8))) float v8f;

__device__ __forceinline__ float warp_sum(float v) {
#pragma unroll
  for (int o = 16; o > 0; o >>= 1) v += __shfl_xor(v, o, 32);
  return v;
}

__device__ __forceinline__ float silu_f(float x) {
  return x / (1.f + __expf(-x));
}

// Zero a small scratch region (used as the "W row" for out-of-range N tiles).
__global__ void zero_kernel(float* __restrict__ p, int n) {
  const int i = blockIdx.x * blockDim.x + threadIdx.x;
  if (i < n) p[i] = 0.f;
}

// ---------------------------------------------------------------------------
// WMMA GEMM: C[m,n] = sum_k A[m,k] * W[n,k]  (+ bias[n]) (+ addsrc[m,n])
// A: (M,K) row-major, W: (N,K) row-major.
// One wave = (16*MT) x (16*NT) tile; 8 waves/block stack in M.
// Layout per ISA 7.12.2:
//   A frag (16x4 f32): lane = (k>>1)*16 + m, vgpr = k&1
//   B frag (4x16 f32): lane = (k>>1)*16 + n, vgpr = k&1
//   C/D (16x16 f32):   vgpr v -> row m0 + v + 8*(lane>>4), col = lane&15
// Out-of-range n rows read `zeros` (>= K+2 floats of 0) -> WMMA no-op,
// keeping EXEC full and the inner loop mask-free.
// ---------------------------------------------------------------------------
template <int NT, int MT>
__global__ __launch_bounds__(256, 1) void gemm_wmma_f32(
    const float* __restrict__ A, const float* __restrict__ W,
    const float* __restrict__ bias, const float* __restrict__ addsrc,
    const float* __restrict__ zeros, float* __restrict__ C,
    int M, int N, int K) {
  const int lane = threadIdx.x & 31;
  const int wv   = threadIdx.x >> 5;
  const int m0   = blockIdx.y * (128 * MT) + wv * (16 * MT);
  const int n0   = blockIdx.x * (16 * NT);
  const int half = lane >> 4;      // 0 or 1
  const int l15  = lane & 15;
  const int kb   = half * 2;       // k sub-offset owned by this lane

  const float* Arow[MT];
#pragma unroll
  for (int s = 0; s < MT; ++s)
    Arow[s] = A + (size_t)(m0 + s * 16 + l15) * K + kb;

  const float* Wrow[NT];
#pragma unroll
  for (int t = 0; t < NT; ++t) {
    const int n = n0 + t * 16 + l15;
    Wrow[t] = (n < N) ? (W + (size_t)n * K + kb) : (zeros + kb);
  }

  __builtin_prefetch(Arow[0], 0, 1);         // global_prefetch_b8
  __builtin_prefetch(Wrow[0], 0, 1);

  v8f acc[MT][NT];
#pragma unroll
  for (int s = 0; s < MT; ++s)
#pragma unroll
    for (int t = 0; t < NT; ++t)
      acc[s][t] = (v8f){0.f,0.f,0.f,0.f,0.f,0.f,0.f,0.f};

#pragma unroll 4
  for (int k = 0; k < K; k += 4) {
    // Load phase: all fragments into distinct registers (up to 6 loads
    // in flight per k-step, 24 across the 4x unroll).
    v2f a[MT], b[NT];
#pragma unroll
    for (int s = 0; s < MT; ++s) a[s] = *(const v2f*)(Arow[s] + k);
#pragma unroll
    for (int t = 0; t < NT; ++t) b[t] = *(const v2f*)(Wrow[t] + k);
    // Compute phase: 8 independent WMMA accumulator chains.
#pragma unroll
    for (int t = 0; t < NT; ++t) {
#pragma unroll
      for (int s = 0; s < MT; ++s) {
        acc[s][t] = __builtin_amdgcn_wmma_f32_16x16x4_f32(
            false, a[s], false, b[t], (short)0, acc[s][t], false, false);
      }
    }
  }

  const int row_half = half * 8;
#pragma unroll
  for (int s = 0; s < MT; ++s) {
#pragma unroll
    for (int t = 0; t < NT; ++t) {
      const int n = n0 + t * 16 + l15;
      if (n >= N) continue;
      const float bv = bias ? bias[n] : 0.f;
#pragma unroll
      for (int v = 0; v < 8; ++v) {
        const int row = m0 + s * 16 + row_half + v;
        const size_t idx = (size_t)row * N + n;
        float val = acc[s][t][v] + bv;
        if (addsrc) val += addsrc[idx];
        C[idx] = val;
      }
    }
  }
}

// ---------------------------------------------------------------------------
// LayerNorm over D=128. One wave per row, 4 elements per lane (float4).
// ---------------------------------------------------------------------------
__global__ __launch_bounds__(256) void layernorm128_kernel(
    const float* __restrict__ x, const float* __restrict__ w,
    const float* __restrict__ b, float* __restrict__ out, int rows) {
  const int lane = threadIdx.x & 31;
  const int row  = blockIdx.x * 8 + (threadIdx.x >> 5);
  if (row >= rows) return;
  const float4 v = *(const float4*)(x + (size_t)row * 128 + lane * 4);
  const float mu = warp_sum(v.x + v.y + v.z + v.w) * (1.f / 128.f);
  const float dx = v.x - mu, dy = v.y - mu, dz = v.z - mu, dw = v.w - mu;
  const float var = warp_sum(dx*dx + dy*dy + dz*dz + dw*dw) * (1.f / 128.f);
  const float inv = rsqrtf(var + 1e-5f);
  const float4 wv = *(const float4*)(w + lane * 4);
  const float4 bv = *(const float4*)(b + lane * 4);
  float4 o;
  o.x = dx * inv * wv.x + bv.x;
  o.y = dy * inv * wv.y + bv.y;
  o.z = dz * inv * wv.z + bv.z;
  o.w = dw * inv * wv.w + bv.w;
  *(float4*)(out + (size_t)row * 128 + lane * 4) = o;
}

// ---------------------------------------------------------------------------
// Depthwise causal conv (K=4) on channels [256,576) of zxbcdt, + bias, SiLU.
// ---------------------------------------------------------------------------
__global__ void conv_silu_kernel(const float* __restrict__ zx,
                                 const float* __restrict__ cw,
                                 const float* __restrict__ cb,
                                 float* __restrict__ xBC, int BL, int L) {
  const int idx = blockIdx.x * blockDim.x + threadIdx.x;
  if (idx >= BL * 320) return;
  const int c = idx % 320;
  const int m = idx / 320;       // m = b*L + l
  const int l = m % L;
  float acc = cb[c];
#pragma unroll
  for (int k = 0; k < 4; ++k) {
    const int lk = l - 3 + k;
    const float xv = (lk >= 0) ? zx[(size_t)(m - 3 + k) * 580 + 256 + c] : 0.f;
    acc += xv * cw[c * 4 + k];
  }
  xBC[(size_t)m * 320 + c] = silu_f(acc);
}

// ---------------------------------------------------------------------------
// dt = softplus(zx[...,576+h] + dt_bias[h]);  dA = exp(-exp(A_log[h]) * dt)
// ---------------------------------------------------------------------------
__global__ void dt_dA_kernel(const float* __restrict__ zx,
                             const float* __restrict__ dt_bias,
                             const float* __restrict__ A_log,
                             float* __restrict__ dt, float* __restrict__ dA,
                             int BL) {
  const int idx = blockIdx.x * blockDim.x + threadIdx.x;
  if (idx >= BL * 4) return;
  const int h = idx & 3;
  const int m = idx >> 2;
  const float v = zx[(size_t)m * 580 + 576 + h] + dt_bias[h];
  const float d = (v > 20.f) ? v : log1pf(__expf(v));
  dt[idx] = d;
  dA[idx] = __expf(-__expf(A_log[h]) * d);
}

// ---------------------------------------------------------------------------
// SSM scan. One wave per (b, head, p) channel; lane n in [0,32) holds h[p,n].
// h = h*dA + (dt*x_p)*B_n ; y_p = sum_n h*C_n (wave reduce) + Dp*x_p.
// 512 independent waves -> fills the machine despite serial L.
// ---------------------------------------------------------------------------
__global__ __launch_bounds__(256) void ssm_scan_kernel(
    const float* __restrict__ xBC, const float* __restrict__ dt,
    const float* __restrict__ dA, const float* __restrict__ Dp,
    float* __restrict__ y, int L, int nwaves) {
  const int lane = threadIdx.x & 31;
  const int wid  = blockIdx.x * 8 + (threadIdx.x >> 5);
  if (wid >= nwaves) return;
  const int b   = wid >> 8;        // 256 channels per batch (4 heads * 64 p)
  const int rem = wid & 255;
  const int hh  = rem >> 6;
  const int p   = rem & 63;

  const float* xb  = xBC + (size_t)b * L * 320;
  const float* dtb = dt + (size_t)b * L * 4 + hh;
  const float* dAb = dA + (size_t)b * L * 4 + hh;
  float* yb = y + (size_t)b * L * 256 + hh * 64 + p;
  const float Dh   = Dp[hh];
  const int   xoff = hh * 64 + p;

  float hn = 0.f;
  for (int t = 0; t < L; ++t) {
    const float* row = xb + (size_t)t * 320;
    const float Bt  = row[256 + lane];
    const float Ct  = row[288 + lane];
    const float xt  = row[xoff];
    const float dAt = dAb[t * 4];
    const float dtt = dtb[t * 4];
    hn = hn * dAt + (dtt * xt) * Bt;
    const float yv = warp_sum(hn * Ct);
    if (lane == 0) yb[(size_t)t * 256] = yv + Dh * xt;
  }
}

// ---------------------------------------------------------------------------
// y = y * silu(z); RMS-norm over 256; * rms_w.  One wave per row.
// ---------------------------------------------------------------------------
__global__ __launch_bounds__(256) void gated_rmsnorm_kernel(
    const float* __restrict__ y, const float* __restrict__ zx,
    const float* __restrict__ rw, float* __restrict__ out, int rows) {
  const int lane = threadIdx.x & 31;
  const int row  = blockIdx.x * 8 + (threadIdx.x >> 5);
  if (row >= rows) return;
  const float* yr = y + (size_t)row * 256;
  const float* zr = zx + (size_t)row * 580;
  float g[8];
  float ss = 0.f;
#pragma unroll
  for (int j = 0; j < 8; ++j) {
    const int c = j * 32 + lane;
    const float z = zr[c];
    const float v = yr[c] * silu_f(z);
    g[j] = v;
    ss += v * v;
  }
  ss = warp_sum(ss);
  const float inv = rsqrtf(ss * (1.f / 256.f) + 1e-5f);
#pragma unroll
  for (int j = 0; j < 8; ++j) {
    const int c = j * 32 + lane;
    out[(size_t)row * 256 + c] = g[j] * inv * rw[c];
  }
}

// ---------------------------------------------------------------------------
// GLU for MLP: h = silu(x12[:, :256]) * x12[:, 256:512]
// ---------------------------------------------------------------------------
__global__ void glu_kernel(const float* __restrict__ x12,
                           float* __restrict__ h, int rows) {
  const int idx = blockIdx.x * blockDim.x + threadIdx.x;
  if (idx >= rows * 256) return;
  const int c = idx & 255;
  const int m = idx >> 8;
  const float a = x12[(size_t)m * 512 + c];
  const float b = x12[(size_t)m * 512 + 256 + c];
  h[idx] = silu_f(a) * b;
}

// ---------------------------------------------------------------------------

static inline int ceil_div(int a, int b) { return (a + b - 1) / b; }

extern "C" void kernel_launch(void* const* d_in, const int* in_sizes, int n_in,
                              void* d_out, int out_size, void* d_ws,
                              size_t ws_size, hipStream_t stream) {
  const float* x_in      = (const float*)d_in[0];
  const float* ln1_w     = (const float*)d_in[2];
  const float* ln1_b     = (const float*)d_in[3];
  const float* ln2_w     = (const float*)d_in[4];
  const float* ln2_b     = (const float*)d_in[5];
  const float* fc1_w     = (const float*)d_in[6];
  const float* fc1_b     = (const float*)d_in[7];
  const float* fc2_w     = (const float*)d_in[8];
  const float* fc2_b     = (const float*)d_in[9];
  const float* in_proj_w = (const float*)d_in[10];
  const float* conv_w    = (const float*)d_in[11];
  const float* conv_b    = (const float*)d_in[12];
  const float* dt_bias   = (const float*)d_in[13];
  const float* A_log     = (const float*)d_in[14];
  const float* Dp        = (const float*)d_in[15];
  const float* rms_w     = (const float*)d_in[16];
  const float* out_proj  = (const float*)d_in[17];

  const int D = 128, DI = 256, CONV = 320, EPROJ = 580;
  const int L = 4096;
  const int BL = in_sizes[0] / D;   // 8192
  const int nb = BL / L;            // 2
  const int NBLK = 8;

  // Workspace partition (floats).
  float* ws = (float*)d_ws;
  float* zbuf    = ws;                 ws += 512;                 // zero rows
  float* x1_buf  = ws;                 ws += (size_t)BL * D;      // LN output
  float* zx_buf  = ws;                 ws += (size_t)BL * EPROJ;  // in_proj out
  float* xBC_buf = ws;                 ws += (size_t)BL * CONV;   // conv+silu
  float* dt_buf  = ws;                 ws += (size_t)BL * 4;
  float* dA_buf  = ws;                 ws += (size_t)BL * 4;
  float* y_buf   = ws;                 ws += (size_t)BL * DI;     // scan out
  float* yn_buf  = ws;                 ws += (size_t)BL * DI;     // gated rms
  float* xm_buf  = ws;                 ws += (size_t)BL * D;      // residual2
  float* x12_buf = ws;                 ws += (size_t)BL * 512;    // fc1 out
  float* h_buf   = ws;                 ws += (size_t)BL * DI;     // glu out
  float* x_buf   = ws;                 ws += (size_t)BL * D;      // block out

  const dim3 blk(256);
  const int gy = BL / 256;             // GEMM M tiles (MT=2 -> 256 rows/block)
  const int nwaves = nb * 256;         // scan channels

  // Zero rows for out-of-range GEMM N tiles (must cover K+2 <= 258 floats).
  zero_kernel<<<2, 256, 0, stream>>>(zbuf, 512);

  const float* xin = x_in;
  for (int i = 0; i < NBLK; ++i) {
    const float* inW  = in_proj_w + (size_t)i * EPROJ * D;
    const float* cW   = conv_w   + (size_t)i * CONV * 4;
    const float* cB   = conv_b   + (size_t)i * CONV;
    const float* dtB  = dt_bias  + (size_t)i * 4;
    const float* Alg  = A_log    + (size_t)i * 4;
    const float* Dpi  = Dp       + (size_t)i * 4;
    const float* rmsW = rms_w    + (size_t)i * DI;
    const float* oW   = out_proj + (size_t)i * D * DI;
    const float* f1W  = fc1_w    + (size_t)i * 512 * D;
    const float* f1B  = fc1_b    + (size_t)i * 512;
    const float* f2W  = fc2_w    + (size_t)i * D * DI;
    const float* f2B  = fc2_b    + (size_t)i * D;

    // x1 = LN(x)
    layernorm128_kernel<<<BL / 8, blk, 0, stream>>>(
        xin, ln1_w + (size_t)i * D, ln1_b + (size_t)i * D, x1_buf, BL);

    // zxbcdt = x1 @ in_proj_w^T  (M=BL, N=580, K=128)
    gemm_wmma_f32<4, 2><<<dim3(ceil_div(EPROJ, 64), gy), blk, 0, stream>>>(
        x1_buf, inW, nullptr, nullptr, zbuf, zx_buf, BL, EPROJ, D);

    // depthwise conv + bias + silu
    conv_silu_kernel<<<ceil_div(BL * CONV, 256), blk, 0, stream>>>(
        zx_buf, cW, cB, xBC_buf, BL, L);

    // dt, dA
    dt_dA_kernel<<<ceil_div(BL * 4, 256), blk, 0, stream>>>(
        zx_buf, dtB, Alg, dt_buf, dA_buf, BL);

    // SSM scan (+ Dp * x skip)
    ssm_scan_kernel<<<ceil_div(nwaves, 8), blk, 0, stream>>>(
        xBC_buf, dt_buf, dA_buf, Dpi, y_buf, L, nwaves);

    // y * silu(z), RMS-norm, * rms_w
    gated_rmsnorm_kernel<<<BL / 8, blk, 0, stream>>>(
        y_buf, zx_buf, rmsW, yn_buf, BL);

    // x_m = xin + yn @ out_proj^T  (N=128, K=256, fused residual add)
    gemm_wmma_f32<4, 2><<<dim3(ceil_div(D, 64), gy), blk, 0, stream>>>(
        yn_buf, oW, nullptr, xin, zbuf, xm_buf, BL, D, DI);

    // x2 = LN(x_m)
    layernorm128_kernel<<<BL / 8, blk, 0, stream>>>(
        xm_buf, ln2_w + (size_t)i * D, ln2_b + (size_t)i * D, x1_buf, BL);

    // x12 = x2 @ fc1^T + b1  (N=512, K=128)
    gemm_wmma_f32<4, 2><<<dim3(ceil_div(512, 64), gy), blk, 0, stream>>>(
        x1_buf, f1W, f1B, nullptr, zbuf, x12_buf, BL, 512, D);

    // h = silu(x1) * x2
    glu_kernel<<<ceil_div(BL * DI, 256), blk, 0, stream>>>(x12_buf, h_buf, BL);

    // x = x_m + h @ fc2^T + b2  (N=128, K=256, fused residual add)
    gemm_wmma_f32<4, 2><<<dim3(ceil_div(D, 64), gy), blk, 0, stream>>>(
        h_buf, f2W, f2B, xm_buf, zbuf, x_buf, BL, D, DI);

    xin = x_buf;
  }

  // Outputs: (x, residual) where residual = x_m of the last block.
  const size_t tensor_bytes = (size_t)BL * D * sizeof(float);
  hipMemcpyAsync(d_out, x_buf, tensor_bytes, hipMemcpyDeviceToDevice, stream);
  hipMemcpyAsync((float*)d_out + (size_t)BL * D, xm_buf, tensor_bytes,
                 hipMemcpyDeviceToDevice, stream);
}